// PatchAttention_72413148610897
// MI455X (gfx1250) — compile-verified
//
#include <hip/hip_runtime.h>
#include <math.h>

// ---------------------------------------------------------------------------
// PatchAttention on MI455X (gfx1250), fp32.
//
// Algebraic restructuring (exact): unfold/PSA/fold per window collapses to
//   B_i = branch_i + pos                       (never materialized; read fused)
//   g[n,s]   = ch_wq_w . B_i[:,s] + b          (per-pixel dot over C)
//   e = exp(g-max); Z = sum_s cnt[s]*e[s]
//   t[c]     = sum_s cnt*B_i[c,s]*e[s]/Z       (channel moment)
//   m[c]     = sum_s cnt*B_i[c,s]              (channel moment)
//   z=Wv@t+b; wz=Wz@z+b; LN; ch_w=sigmoid(...)  (tiny head)
//   sq=softmax(Wq@(m/J)+b); qt=sq@Wv2; beta=sq.bv
//   sigw[n,s]= sigmoid(qt . B_i[:,s] + beta)   (per-pixel dot over C)
//   G_i      = B_i*(1+ch_w+sigw) - pos         (fused into res-GEMM B staging)
// final: out = Wres@G + b + x ; y = Wfus@out + b
//
// GEMMs: V_WMMA_F32_16X16X4_F32 (native fp32 matrix path; reference is fp32).
// B tiles are staged once per block in LDS (K-pairs as float2, pair-row
// stride 80 -> lanes 0-15 / 16-31 hit disjoint bank halves, conflict-free
// ds_load_b64 fragments). This removes the 8x per-wave redundant global loads
// the previous version had. Problem is HBM-bound (~500 MB @ 23.3 TB/s).
// ---------------------------------------------------------------------------

typedef float v2f __attribute__((ext_vector_type(2)));
typedef float v8f __attribute__((ext_vector_type(8)));

#define NB   8
#define CIN  256
#define HWSZ 4096
#define OEXP 768

// LDS B-tile: 16 k-rows x 64 m-cols stored as 8 pair-rows of float2.
// float index of element (k, m):
#define BS_IDX(k, m) (((k) >> 1) * 160 + ((m) << 1) + ((k) & 1))

// ---- window geometry -------------------------------------------------------
__device__ __forceinline__ int cnt1d(int h, int sh, int kh, int oh) {
  int amax = h / sh; if (amax > oh - 1) amax = oh - 1;
  int t = h - kh + 1;
  int amin = (t <= 0) ? 0 : (t + sh - 1) / sh;
  return amax - amin + 1;
}
__device__ __forceinline__ float cntf(int i, int s) {
  int h = s >> 6, w = s & 63;
  int sh, kh, oh;
  if (i == 0)      { sh = 32; kh = 32; oh = 2; }
  else if (i == 1) { sh = 16; kh = 16; oh = 4; }
  else             { sh = 10; kh = 14; oh = 6; }
  return (float)(cnt1d(h, sh, kh, oh) * cnt1d(w, sh, kh, oh));
}

// ---- block reductions ------------------------------------------------------
__device__ __forceinline__ float block_sum(float v, float* red) {
  int tid = threadIdx.x;
  red[tid] = v; __syncthreads();
  for (int off = 128; off > 0; off >>= 1) {
    if (tid < off) red[tid] += red[tid + off];
    __syncthreads();
  }
  float r = red[0]; __syncthreads();
  return r;
}
__device__ __forceinline__ float block_max(float v, float* red) {
  int tid = threadIdx.x;
  red[tid] = v; __syncthreads();
  for (int off = 128; off > 0; off >>= 1) {
    if (tid < off) red[tid] = fmaxf(red[tid], red[tid + off]);
    __syncthreads();
  }
  float r = red[0]; __syncthreads();
  return r;
}

// ---- positional encoding (replicates the reference's raw reshape) ----------
__global__ __launch_bounds__(256) void pos_kernel(float* __restrict__ pos) {
  int idx = blockIdx.x * 256 + threadIdx.x;          // < 256*4096
  int s   = idx & 4095;
  int cp  = idx >> 12;
  int c   = s & 255;                                  // pe column
  int r   = (cp << 4) + (s >> 8);                     // pe row
  int iw  = r >> 6, ih = r & 63;                      // meshgrid 'ij': (W,H)
  int q   = c >> 6, k = c & 63;
  float omega = expf(-(float)k * (9.210340371976184f / 64.0f)); // 10000^(-k/64)
  float arg = (q < 2 ? (float)iw : (float)ih) * omega;
  pos[idx] = ((q & 1) == 0) ? sinf(arg) : cosf(arg);
}

// ---- WMMA fp32 GEMM: Out[n,o,m] = W[o,:]@B[:,m] + bias (+residual) ---------
// MODE 0: B = In[n,c,m]     MODE 1: B = (expx+pos)*(1+chw+sigw) - pos (fused G)
// Block: 256 thr = 8 waves. Tile 128(O) x 64(M); wave w owns rows [16w,16w+16).
// B tile staged in LDS per 16-k chunk; fragments per ISA 7.12.2 layouts
// (A 16x4: lane=row, K pair by lane[4]; B 4x16 / D 16x16: rows across lanes).
template <int MODE>
__global__ __launch_bounds__(256)
void gemm1x1(const float* __restrict__ W, const float* __restrict__ bias,
             const float* __restrict__ In, float* __restrict__ Out,
             const float* __restrict__ residual, int O, int Cin,
             const float* __restrict__ pos, const float* __restrict__ chw,
             const float* __restrict__ sigw) {
  __shared__ float Bs[8 * 160];          // 16k x 64m as float2 pairs, 5 KB
  const int n    = blockIdx.z;
  const int oblk = blockIdx.y * 128;
  const int mblk = blockIdx.x * 64;
  const int tid  = threadIdx.x;
  const int wave = tid >> 5;
  const int lane = tid & 31;
  const int lhi  = lane >> 4;       // selects K pair (A/B) & row half (D)
  const int llo  = lane & 15;

  v8f acc[4];
#pragma unroll
  for (int mt = 0; mt < 4; ++mt)
#pragma unroll
    for (int r = 0; r < 8; ++r) acc[mt][r] = 0.0f;

  const int    orow = oblk + wave * 16 + llo;
  const float* Wrow = W + (size_t)orow * Cin;
  const int    fm   = tid & 63;     // cooperative-fill column
  const int    fk   = tid >> 6;     // cooperative-fill row base (stride 4)
  const int    gm   = mblk + fm;

  for (int kb = 0; kb < Cin; kb += 16) {
    // ---- cooperative fill of B tile (each element computed once) ----
#pragma unroll
    for (int j = 0; j < 4; ++j) {
      const int k = fk + 4 * j;
      const int c = kb + k;
      float v;
      if (MODE == 0) {
        v = In[((size_t)n * Cin + c) * HWSZ + gm];
      } else {
        const int i = c >> 8, cc = c & 255;
        const float p = pos[(size_t)cc * HWSZ + gm];
        const float f = 1.0f + chw[(i * NB + n) * 256 + cc]
                             + sigw[(size_t)(i * NB + n) * HWSZ + gm];
        v = (In[((size_t)n * OEXP + c) * HWSZ + gm] + p) * f - p;
      }
      Bs[BS_IDX(k, fm)] = v;
    }
    __syncthreads();
    // ---- 4 WMMA k-steps from LDS ----
#pragma unroll
    for (int ks = 0; ks < 16; ks += 4) {
      const int klo = ks + lhi * 2;
      v2f a = *(const v2f*)(Wrow + kb + klo);          // A[row, klo..klo+1]
#pragma unroll
      for (int mt = 0; mt < 4; ++mt) {
        v2f b = *(const v2f*)&Bs[BS_IDX(klo, mt * 16 + llo)];
        acc[mt] = __builtin_amdgcn_wmma_f32_16x16x4_f32(
            false, a, false, b, (short)0, acc[mt], false, false);
      }
    }
    __syncthreads();
  }

#pragma unroll
  for (int mt = 0; mt < 4; ++mt) {
    const int m = mblk + mt * 16 + llo;
#pragma unroll
    for (int r = 0; r < 8; ++r) {
      const int o = oblk + wave * 16 + r + lhi * 8;   // D row mapping
      float v = acc[mt][r] + bias[o];
      if (MODE == 1) v += residual[((size_t)n * O + o) * HWSZ + m];
      Out[((size_t)n * O + o) * HWSZ + m] = v;
    }
  }
}

// ---- per-pixel dot over C: out = [sigmoid]( wvec . B_i[:,s] + add ) --------
__global__ __launch_bounds__(256)
void pixel_dot(const float* __restrict__ expx, const float* __restrict__ pos,
               const float* __restrict__ wbase, int wstride,
               const float* __restrict__ abase, int astride,
               int do_sig, float* __restrict__ outb) {
  const int i = blockIdx.y, n = blockIdx.z;
  const int in = i * NB + n;
  const int s = blockIdx.x * 256 + threadIdx.x;
  const float* E  = expx + ((size_t)(n * 3 + i) * 256) * HWSZ;
  const float* wv = wbase + (size_t)in * wstride;
  float acc = abase[in * astride];
  for (int c = 0; c < 256; ++c)
    acc = fmaf(E[(size_t)c * HWSZ + s] + pos[(size_t)c * HWSZ + s], wv[c], acc);
  if (do_sig) acc = 1.0f / (1.0f + expf(-acc));
  outb[(size_t)in * HWSZ + s] = acc;
}

// ---- cnt-weighted softmax normalizer over pixels; g -> exp(g-max) ----------
__global__ __launch_bounds__(256)
void softmax_z(float* __restrict__ g, float* __restrict__ Z) {
  __shared__ float red[256];
  const int in = blockIdx.x;           // i*NB + n
  const int i  = in >> 3;
  float* G = g + (size_t)in * HWSZ;
  float mx = -3.4e38f;
  for (int s = threadIdx.x; s < HWSZ; s += 256) mx = fmaxf(mx, G[s]);
  mx = block_max(mx, red);
  float sum = 0.0f;
  for (int s = threadIdx.x; s < HWSZ; s += 256) {
    float e = expf(G[s] - mx);
    G[s] = e;
    sum += cntf(i, s) * e;
  }
  sum = block_sum(sum, red);
  if (threadIdx.x == 0) Z[in] = sum;
}

// ---- channel moments: m[c]=sum cnt*B, t[c]=sum cnt*B*e/Z -------------------
__global__ __launch_bounds__(256)
void moment_kernel(const float* __restrict__ expx, const float* __restrict__ pos,
                   const float* __restrict__ ebuf, const float* __restrict__ Z,
                   float* __restrict__ mo, float* __restrict__ to) {
  __shared__ float red[256];
  const int c = blockIdx.x, i = blockIdx.y, n = blockIdx.z;
  const int in = i * NB + n;
  const float* row = expx + ((size_t)(n * 3 + i) * 256 + c) * HWSZ;
  const float* pr  = pos + (size_t)c * HWSZ;
  const float* E   = ebuf + (size_t)in * HWSZ;
  float am = 0.0f, at = 0.0f;
  for (int s = threadIdx.x; s < HWSZ; s += 256) {
    float b = row[s] + pr[s];
    float cn = cntf(i, s);
    am = fmaf(cn, b, am);
    at = fmaf(cn * b, E[s], at);
  }
  am = block_sum(am, red);
  at = block_sum(at, red);
  if (threadIdx.x == 0) {
    mo[in * 256 + c] = am;
    to[in * 256 + c] = at / Z[in];
  }
}

// ---- tiny per-(window,n) head: z, wz, LN->ch_w, sq softmax, qt, beta -------
__global__ __launch_bounds__(256)
void head_kernel(const float* __restrict__ mbuf, const float* __restrict__ tbuf,
                 const float* __restrict__ wv_w, const float* __restrict__ wv_b,
                 const float* __restrict__ wz_w, const float* __restrict__ wz_b,
                 const float* __restrict__ ln_g, const float* __restrict__ ln_b,
                 const float* __restrict__ sq_w, const float* __restrict__ sq_b,
                 const float* __restrict__ sv_w, const float* __restrict__ sv_b,
                 float* __restrict__ chw, float* __restrict__ qt,
                 float* __restrict__ beta) {
  __shared__ float t_l[256], m_l[256], z_l[128], s_l[128], red[256];
  const int in  = blockIdx.x;
  const int i   = in >> 3;
  const int tid = threadIdx.x;
  const float invJ = (i == 2) ? (1.0f / 7056.0f) : (1.0f / 4096.0f);
  t_l[tid] = tbuf[in * 256 + tid];
  m_l[tid] = mbuf[in * 256 + tid] * invJ;
  __syncthreads();
  if (tid < 128) {                                     // z = Wv@t + b (128)
    float a = wv_b[tid];
    for (int c = 0; c < 256; ++c) a = fmaf(wv_w[tid * 256 + c], t_l[c], a);
    z_l[tid] = a;
  }
  __syncthreads();
  float wz = wz_b[tid];                                // wz = Wz@z + b (256)
  for (int c = 0; c < 128; ++c) wz = fmaf(wz_w[tid * 128 + c], z_l[c], wz);
  float mean = block_sum(wz, red) * (1.0f / 256.0f);   // LayerNorm + sigmoid
  float d = wz - mean;
  float var = block_sum(d * d, red) * (1.0f / 256.0f);
  float lnv = d * rsqrtf(var + 1e-5f) * ln_g[tid] + ln_b[tid];
  chw[in * 256 + tid] = 1.0f / (1.0f + expf(-lnv));
  if (tid < 128) {                                     // sq logits (128)
    float a = sq_b[tid];
    for (int c = 0; c < 256; ++c) a = fmaf(sq_w[tid * 256 + c], m_l[c], a);
    s_l[tid] = a;
  }
  __syncthreads();
  float sv = (tid < 128) ? s_l[tid] : -3.4e38f;        // softmax over 128
  float mx = block_max(sv, red);
  float ev = (tid < 128) ? expf(s_l[tid] - mx) : 0.0f;
  float ss = block_sum(ev, red);
  if (tid < 128) s_l[tid] = ev / ss;
  __syncthreads();
  float a = 0.0f;                                      // qt = sq @ sp_wv_w
  for (int c = 0; c < 128; ++c) a = fmaf(s_l[c], sv_w[c * 256 + tid], a);
  qt[in * 256 + tid] = a;
  float bv = (tid < 128) ? s_l[tid] * sv_b[tid] : 0.0f;
  bv = block_sum(bv, red);
  if (tid == 0) beta[in] = bv;
}

// ---------------------------------------------------------------------------
extern "C" void kernel_launch(void* const* d_in, const int* in_sizes, int n_in,
                              void* d_out, int out_size, void* d_ws, size_t ws_size,
                              hipStream_t stream) {
  (void)in_sizes; (void)n_in; (void)out_size; (void)ws_size;
  const float* x       = (const float*)d_in[0];
  const float* w_exp   = (const float*)d_in[1];
  const float* b_exp   = (const float*)d_in[2];
  const float* w_res   = (const float*)d_in[3];
  const float* b_res   = (const float*)d_in[4];
  const float* w_fus   = (const float*)d_in[5];
  const float* b_fus   = (const float*)d_in[6];
  const float* ch_wv_w = (const float*)d_in[7];
  const float* ch_wv_b = (const float*)d_in[8];
  const float* ch_wq_w = (const float*)d_in[9];
  const float* ch_wq_b = (const float*)d_in[10];
  const float* ch_wz_w = (const float*)d_in[11];
  const float* ch_wz_b = (const float*)d_in[12];
  const float* ln_g    = (const float*)d_in[13];
  const float* ln_b    = (const float*)d_in[14];
  const float* sp_wv_w = (const float*)d_in[15];
  const float* sp_wv_b = (const float*)d_in[16];
  const float* sp_wq_w = (const float*)d_in[17];
  const float* sp_wq_b = (const float*)d_in[18];

  float* ws = (float*)d_ws;
  size_t o = 0;
  float* pos  = ws + o; o += (size_t)CIN * HWSZ;        // 4 MB
  float* expx = ws + o; o += (size_t)NB * OEXP * HWSZ;  // 100 MB
  float* gbuf = ws + o; o += (size_t)24 * HWSZ;
  float* sigw = ws + o; o += (size_t)24 * HWSZ;
  float* mbuf = ws + o; o += 24 * 256;
  float* tbuf = ws + o; o += 24 * 256;
  float* chw  = ws + o; o += 24 * 256;
  float* qt   = ws + o; o += 24 * 256;
  float* Zb   = ws + o; o += 32;
  float* beta = ws + o; o += 32;
  float* outb = ws + o;                                 // 33.5 MB

  pos_kernel<<<dim3((CIN * HWSZ) / 256), 256, 0, stream>>>(pos);

  gemm1x1<0><<<dim3(HWSZ / 64, OEXP / 128, NB), 256, 0, stream>>>(
      w_exp, b_exp, x, expx, nullptr, OEXP, CIN, nullptr, nullptr, nullptr);

  pixel_dot<<<dim3(HWSZ / 256, 3, NB), 256, 0, stream>>>(
      expx, pos, ch_wq_w, 0, ch_wq_b, 0, 0, gbuf);

  softmax_z<<<dim3(24), 256, 0, stream>>>(gbuf, Zb);

  moment_kernel<<<dim3(256, 3, NB), 256, 0, stream>>>(
      expx, pos, gbuf, Zb, mbuf, tbuf);

  head_kernel<<<dim3(24), 256, 0, stream>>>(
      mbuf, tbuf, ch_wv_w, ch_wv_b, ch_wz_w, ch_wz_b, ln_g, ln_b,
      sp_wq_w, sp_wq_b, sp_wv_w, sp_wv_b, chw, qt, beta);

  pixel_dot<<<dim3(HWSZ / 256, 3, NB), 256, 0, stream>>>(
      expx, pos, qt, 256, beta, 1, 1, sigw);

  gemm1x1<1><<<dim3(HWSZ / 64, 256 / 128, NB), 256, 0, stream>>>(
      w_res, b_res, expx, outb, x, 256, OEXP, pos, chw, sigw);

  gemm1x1<0><<<dim3(HWSZ / 64, 256 / 128, NB), 256, 0, stream>>>(
      w_fus, b_fus, outb, (float*)d_out, nullptr, 256, CIN,
      nullptr, nullptr, nullptr);
}